// GL_MoCo_70257075028420
// MI455X (gfx1250) — compile-verified
//
#include <hip/hip_runtime.h>

// GL-MoCo forward loss on gfx1250 (MI455X), fp32 WMMA path.
// Shapes fixed by the reference: B=512, K=8192, D=128, N=B+K=8704, knn=20, T=0.1.

#define DIM    128
#define BQ     512
#define KQ     8192
#define NNODE  8704          // BQ + KQ
#define KNN    20
#define ROWBLK 512           // rows of the distance matrix processed per block
#define INVT   10.0f         // 1/T
#define DNCONS 0.22360679774997896f  // in_deg == 20 for every node -> 20^{-1/2}

typedef __attribute__((ext_vector_type(2))) float v2f;
typedef __attribute__((ext_vector_type(8))) float v8f;

static __device__ __forceinline__ v8f wmma4(v2f a, v2f b, v8f c) {
  // V_WMMA_F32_16X16X4_F32 : D(16x16 f32) = A(16x4 f32) * B(4x16 f32) + C
  return __builtin_amdgcn_wmma_f32_16x16x4_f32(false, a, false, b, (short)0, c,
                                               false, false);
}

// ---------------------------------------------------------------- normalize
// Row l2-norm (eps 1e-12), 1 block per row, 128 threads.
__global__ void k_rownorm(const float* __restrict__ in, float* __restrict__ out) {
  __shared__ float r[DIM];
  const int i = blockIdx.x, d = threadIdx.x;
  float x = in[(size_t)i * DIM + d];
  r[d] = x * x;
  __syncthreads();
  for (int s = 64; s > 0; s >>= 1) {
    if (d < s) r[d] += r[d + s];
    __syncthreads();
  }
  float rn = 1.0f / fmaxf(sqrtf(r[0]), 1e-12f);
  out[(size_t)i * DIM + d] = x * rn;
}

// Column-normalize queue [D,K]; also scatter queue_n^T rows into Xq and Xk.
__global__ void k_colnorm(const float* __restrict__ queue, float* __restrict__ qn,
                          float* __restrict__ Xq, float* __restrict__ Xk) {
  const int j = blockIdx.x * blockDim.x + threadIdx.x;  // 0..KQ-1
  float ss = 0.0f;
  for (int d = 0; d < DIM; ++d) {
    float v = queue[(size_t)d * KQ + j];
    ss += v * v;
  }
  float rn = 1.0f / fmaxf(sqrtf(ss), 1e-12f);
  for (int d = 0; d < DIM; ++d) {
    float v = queue[(size_t)d * KQ + j] * rn;
    qn[(size_t)d * KQ + j] = v;
    Xq[(size_t)(BQ + j) * DIM + d] = v;
    Xk[(size_t)(BQ + j) * DIM + d] = v;
  }
}

// Per-row softmax over DIM features + l2-norm of the softmax row.
__global__ void k_softmax(const float* __restrict__ X, float* __restrict__ S,
                          float* __restrict__ w) {
  __shared__ float r[DIM];
  const int i = blockIdx.x, d = threadIdx.x;
  float x = X[(size_t)i * DIM + d];
  r[d] = x;
  __syncthreads();
  for (int s = 64; s > 0; s >>= 1) {
    if (d < s) r[d] = fmaxf(r[d], r[d + s]);
    __syncthreads();
  }
  float m = r[0];
  __syncthreads();
  float e = expf(x - m);
  r[d] = e;
  __syncthreads();
  for (int s = 64; s > 0; s >>= 1) {
    if (d < s) r[d] += r[d + s];
    __syncthreads();
  }
  float sv = e / r[0];
  __syncthreads();
  S[(size_t)i * DIM + d] = sv;
  r[d] = sv * sv;
  __syncthreads();
  for (int s = 64; s > 0; s >>= 1) {
    if (d < s) r[d] += r[d + s];
    __syncthreads();
  }
  if (d == 0) w[i] = sqrtf(r[0]);
}

// ------------------------------------------------- distance GEMM (WMMA f32)
// dist[i][j] = 1 - (S_i . S_j)/max(w_i*w_j, 1e-7), diagonal forced to -1.
// One wave -> one 16x16 tile; ROWBLK rows x NNODE cols per launch.
__global__ void k_dist_gemm(const float* __restrict__ S, const float* __restrict__ w,
                            float* __restrict__ dist, int row0base) {
  const int wave = threadIdx.x >> 5;
  const int lane = threadIdx.x & 31;
  const int tile = blockIdx.x * 4 + wave;
  const int colTiles = NNODE / 16;                 // 544
  const int rt = tile / colTiles;                  // 0..ROWBLK/16-1
  const int ct = tile % colTiles;
  const int row0 = row0base + rt * 16;
  const int col0 = ct * 16;
  const int half = lane >> 4, l = lane & 15;

  const float* __restrict__ Arow = S + (size_t)(row0 + l) * DIM;
  const float* __restrict__ Brow = S + (size_t)(col0 + l) * DIM;

  v8f c = {};
  for (int k0 = 0; k0 < DIM; k0 += 4) {
    v2f a, b;
    a.x = Arow[k0 + 2 * half];
    a.y = Arow[k0 + 2 * half + 1];
    b.x = Brow[k0 + 2 * half];
    b.y = Brow[k0 + 2 * half + 1];
    c = wmma4(a, b, c);
  }
  const int j = col0 + l;
  const float wj = w[j];
  for (int v = 0; v < 8; ++v) {
    int m = row0 + v + 8 * half;                    // global row
    float den = fmaxf(w[m] * wj, 1e-7f);
    float dv = 1.0f - c[v] / den;
    dv = (m == j) ? -1.0f : dv;
    dist[(size_t)(m - row0base) * NNODE + j] = dv;
  }
}

// Top-KNN smallest per row (iterative selection, ties -> lowest index).
__global__ void k_topk(float* __restrict__ dist, int* __restrict__ knn, int rowbase) {
  __shared__ float sv[256];
  __shared__ int si[256];
  const int i = rowbase + blockIdx.x;
  float* row = dist + (size_t)blockIdx.x * NNODE;
  const int tid = threadIdx.x;
  for (int t = 0; t < KNN; ++t) {
    float bv = 3.0e38f;
    int bi = NNODE;
    for (int j = tid; j < NNODE; j += 256) {
      float v = row[j];
      if (v < bv || (v == bv && j < bi)) { bv = v; bi = j; }
    }
    sv[tid] = bv; si[tid] = bi;
    __syncthreads();
    for (int s = 128; s > 0; s >>= 1) {
      if (tid < s) {
        float ov = sv[tid + s]; int oi = si[tid + s];
        if (ov < sv[tid] || (ov == sv[tid] && oi < si[tid])) { sv[tid] = ov; si[tid] = oi; }
      }
      __syncthreads();
    }
    if (tid == 0) {
      knn[(size_t)i * KNN + t] = si[0];
      row[si[0]] = 3.0e38f;  // exclude from next pass
    }
    __syncthreads();
  }
}

// out-degree over all edges (src = knn entries).
__global__ void k_deg(const int* __restrict__ knn, float* __restrict__ outdeg) {
  int e = blockIdx.x * blockDim.x + threadIdx.x;   // 0..N*KNN-1
  atomicAdd(&outdeg[knn[e]], 1.0f);
}

__global__ void k_sn(const float* __restrict__ outdeg, float* __restrict__ sn) {
  int i = blockIdx.x * blockDim.x + threadIdx.x;
  sn[i] = rsqrtf(fmaxf(outdeg[i], 1.0f));
}

// agg[i][d] = dn * sum_t relu?(h[knn[i][t]][d]) * sn[knn[i][t]]
__global__ void k_agg(const float* __restrict__ h, const int* __restrict__ knn,
                      const float* __restrict__ sn, float* __restrict__ pre, int relu) {
  const int i = blockIdx.x, d = threadIdx.x;
  float acc = 0.0f;
  for (int t = 0; t < KNN; ++t) {
    int s = knn[(size_t)i * KNN + t];
    float v = h[(size_t)s * DIM + d];
    if (relu) v = fmaxf(v, 0.0f);
    acc += v * sn[s];
  }
  pre[(size_t)i * DIM + d] = acc * DNCONS;
}

// ------------------------------------------------- dense GEMM + bias (WMMA)
// Out[M,Ncols] = A[M,DIM] @ Wm[DIM,Ncols] (+ bias). One wave per 16x16 tile.
__global__ void k_gemm_bias(const float* __restrict__ A, const float* __restrict__ Wm,
                            const float* __restrict__ bias, float* __restrict__ Out,
                            int M, int Ncols) {
  const int wave = threadIdx.x >> 5;
  const int lane = threadIdx.x & 31;
  const int tile = blockIdx.x * 4 + wave;
  const int colTiles = Ncols / 16;
  const int rt = tile / colTiles;
  const int ct = tile % colTiles;
  const int row0 = rt * 16;
  const int col0 = ct * 16;
  const int half = lane >> 4, l = lane & 15;

  const float* __restrict__ Arow = A + (size_t)(row0 + l) * DIM;

  v8f c = {};
  for (int k0 = 0; k0 < DIM; k0 += 4) {
    v2f a, b;
    a.x = Arow[k0 + 2 * half];
    a.y = Arow[k0 + 2 * half + 1];
    b.x = Wm[(size_t)(k0 + 2 * half) * Ncols + col0 + l];
    b.y = Wm[(size_t)(k0 + 2 * half + 1) * Ncols + col0 + l];
    c = wmma4(a, b, c);
  }
  float bv = bias ? bias[col0 + l] : 0.0f;
  for (int v = 0; v < 8; ++v) {
    int m = row0 + v + 8 * half;
    Out[(size_t)m * Ncols + col0 + l] = c[v] + bv;
  }
  (void)M;
}

// ------------------------------------------------- cross-entropy (label 0)
// loss_i = logsumexp([l_pos, l_neg]/T) - l_pos/T ; atomic accumulate.
__global__ void k_ce(const float* __restrict__ Lbuf, const float* __restrict__ A,
                     const float* __restrict__ Bm, float* __restrict__ acc) {
  __shared__ float red[256];
  const int i = blockIdx.x, tid = threadIdx.x;
  // l_pos
  float p = 0.0f;
  for (int d = tid; d < DIM; d += 256) p += A[(size_t)i * DIM + d] * Bm[(size_t)i * DIM + d];
  red[tid] = p;
  __syncthreads();
  for (int s = 128; s > 0; s >>= 1) {
    if (tid < s) red[tid] += red[tid + s];
    __syncthreads();
  }
  float z0 = red[0] * INVT;
  __syncthreads();
  // max
  float m = z0;
  for (int j = tid; j < KQ; j += 256) m = fmaxf(m, Lbuf[(size_t)i * KQ + j] * INVT);
  red[tid] = m;
  __syncthreads();
  for (int s = 128; s > 0; s >>= 1) {
    if (tid < s) red[tid] = fmaxf(red[tid], red[tid + s]);
    __syncthreads();
  }
  m = red[0];
  __syncthreads();
  // sum exp
  float sum = (tid == 0) ? expf(z0 - m) : 0.0f;
  for (int j = tid; j < KQ; j += 256) sum += expf(Lbuf[(size_t)i * KQ + j] * INVT - m);
  red[tid] = sum;
  __syncthreads();
  for (int s = 128; s > 0; s >>= 1) {
    if (tid < s) red[tid] += red[tid + s];
    __syncthreads();
  }
  if (tid == 0) atomicAdd(acc, m + logf(red[0]) - z0);
}

__global__ void k_fin(const float* __restrict__ acc, float* __restrict__ out) {
  out[0] = (acc[0] + acc[1]) * (1.0f / (float)BQ);
}

// ------------------------------------------------- host-side graph pipeline
static void run_graph(const float* X, const float* W1, const float* b1,
                      const float* W2, const float* b2, float* S, float* wv,
                      float* distblk, int* knn, float* outdeg, float* snv,
                      float* pre, float* h1, float* h2, float* gout,
                      hipStream_t stream) {
  k_softmax<<<NNODE, 128, 0, stream>>>(X, S, wv);
  hipMemsetAsync(outdeg, 0, NNODE * sizeof(float), stream);
  for (int rb = 0; rb < NNODE; rb += ROWBLK) {
    k_dist_gemm<<<(ROWBLK / 16) * (NNODE / 16) / 4, 128, 0, stream>>>(S, wv, distblk, rb);
    k_topk<<<ROWBLK, 256, 0, stream>>>(distblk, knn, rb);
  }
  k_deg<<<NNODE * KNN / 256, 256, 0, stream>>>(knn, outdeg);
  k_sn<<<NNODE / 256, 256, 0, stream>>>(outdeg, snv);
  // layer 1
  k_agg<<<NNODE, 128, 0, stream>>>(X, knn, snv, pre, 0);
  k_gemm_bias<<<(NNODE / 16) * (DIM / 16) / 4, 128, 0, stream>>>(pre, W1, b1, h1, NNODE, DIM);
  // layer 2 (relu folded into gather)
  k_agg<<<NNODE, 128, 0, stream>>>(h1, knn, snv, pre, 1);
  k_gemm_bias<<<(NNODE / 16) * (DIM / 16) / 4, 128, 0, stream>>>(pre, W2, b2, h2, NNODE, DIM);
  // final row l2-norm of first BQ rows
  k_rownorm<<<BQ, 128, 0, stream>>>(h2, gout);
}

extern "C" void kernel_launch(void* const* d_in, const int* in_sizes, int n_in,
                              void* d_out, int out_size, void* d_ws, size_t ws_size,
                              hipStream_t stream) {
  (void)in_sizes; (void)n_in; (void)out_size; (void)ws_size;
  const float* im_q  = (const float*)d_in[0];
  const float* im_k  = (const float*)d_in[1];
  const float* queue = (const float*)d_in[2];
  const float* Wq1 = (const float*)d_in[3];
  const float* bq1 = (const float*)d_in[4];
  const float* Wq2 = (const float*)d_in[5];
  const float* bq2 = (const float*)d_in[6];
  const float* Wk1 = (const float*)d_in[7];
  const float* bk1 = (const float*)d_in[8];
  const float* Wk2 = (const float*)d_in[9];
  const float* bk2 = (const float*)d_in[10];
  // d_in[11] = adj_k (==20, compiled in as KNN)

  // -------- workspace carve (all fp32 unless noted)
  float* Xq      = (float*)d_ws;                         // N*D
  float* Xk      = Xq + (size_t)NNODE * DIM;             // N*D
  float* qn      = Xk + (size_t)NNODE * DIM;             // D*K
  float* S       = qn + (size_t)DIM * KQ;                // N*D
  float* wv      = S + (size_t)NNODE * DIM;              // N
  float* distblk = wv + NNODE;                           // ROWBLK*N
  int*   knn     = (int*)(distblk + (size_t)ROWBLK * NNODE);  // N*KNN
  float* outdeg  = (float*)(knn + (size_t)NNODE * KNN);  // N
  float* snv     = outdeg + NNODE;                       // N
  float* pre     = snv + NNODE;                          // N*D
  float* h1      = pre + (size_t)NNODE * DIM;            // N*D
  float* h2      = h1 + (size_t)NNODE * DIM;             // N*D
  float* gq      = h2 + (size_t)NNODE * DIM;             // B*D
  float* gk      = gq + (size_t)BQ * DIM;                // B*D
  float* Lbuf    = gk + (size_t)BQ * DIM;                // B*K
  float* acc     = Lbuf + (size_t)BQ * KQ;               // 2 (its, gts)

  hipMemsetAsync(acc, 0, 2 * sizeof(float), stream);

  // normalize inputs
  k_rownorm<<<BQ, 128, 0, stream>>>(im_q, Xq);
  k_rownorm<<<BQ, 128, 0, stream>>>(im_k, Xk);
  k_colnorm<<<KQ / 256, 256, 0, stream>>>(queue, qn, Xq, Xk);

  // instance-level loss: logits = [q.k, q @ queue_n]/T
  k_gemm_bias<<<(BQ / 16) * (KQ / 16) / 4, 128, 0, stream>>>(Xq, qn, nullptr, Lbuf, BQ, KQ);
  k_ce<<<BQ, 256, 0, stream>>>(Lbuf, Xq, Xk, acc + 0);

  // graph-level: q-graph then k-graph (buffers reused)
  run_graph(Xq, Wq1, bq1, Wq2, bq2, S, wv, distblk, knn, outdeg, snv, pre, h1, h2, gq, stream);
  run_graph(Xk, Wk1, bk1, Wk2, bk2, S, wv, distblk, knn, outdeg, snv, pre, h1, h2, gk, stream);

  // graph-level loss
  k_gemm_bias<<<(BQ / 16) * (KQ / 16) / 4, 128, 0, stream>>>(gq, qn, nullptr, Lbuf, BQ, KQ);
  k_ce<<<BQ, 256, 0, stream>>>(Lbuf, gq, gk, acc + 1);

  k_fin<<<1, 1, 0, stream>>>(acc, (float*)d_out);
}